// Cube2Equirec_1726576857110
// MI455X (gfx1250) — compile-verified
//
#include <hip/hip_runtime.h>
#include <cstdint>

// Problem constants (must match reference)
#define CUBE_L 512
#define EQU_H  512
#define EQU_W  1024
#define NCH    32
#define NE     2

typedef __attribute__((ext_vector_type(4))) unsigned int u32x4;
typedef __attribute__((ext_vector_type(8))) unsigned int u32x8;

// Issue one 2D-tile Tensor-Data-Mover load (global -> LDS), 4-byte elements.
// D# group0 (4 SGPRs) + group1 (8 SGPRs) built per CDNA5 ISA §8 (CDNA5 ISA).
// 2-group form => VADDR2/VADDR3 NULL => tensors up to 2D.
__device__ __forceinline__ void tdm_load_tile_2d(uint32_t lds_addr, uint64_t gaddr,
                                                 uint32_t tensor_dim0, uint32_t tensor_dim1,
                                                 uint32_t dim0_stride,
                                                 uint32_t tile_dim0, uint32_t tile_dim1) {
    u32x4 g0;
    g0[0] = 1u;                                        // count=1, user mode, no gather
    g0[1] = lds_addr;                                  // lds_addr [63:32]
    g0[2] = (uint32_t)(gaddr & 0xFFFFFFFFu);           // global_addr [95:64]
    g0[3] = (uint32_t)(gaddr >> 32) | (2u << 30);      // global_addr hi + type=2 ("image")

    u32x8 g1;
    g1[0] = 2u << 16;                                  // workgroup_mask=0, data_size=2 (4B)
    g1[1] = (tensor_dim0 & 0xFFFFu) << 16;             // tensor_dim0[15:0] @ bits 63:48
    g1[2] = (tensor_dim0 >> 16) |                      // tensor_dim0[31:16]
            ((tensor_dim1 & 0xFFFFu) << 16);           // tensor_dim1[15:0]
    g1[3] = (tensor_dim1 >> 16) |                      // tensor_dim1[31:16]
            (tile_dim0 << 16);                         // tile_dim0 @ bits 127:112
    g1[4] = tile_dim1 & 0xFFFFu;                       // tile_dim1; tile_dim2=0 (2D)
    g1[5] = dim0_stride;                               // tensor_dim0_stride[31:0]
    g1[6] = 0u;                                        // stride0 hi16 = 0, stride1 lo16 = 0
    g1[7] = 0u;                                        // stride1 hi = 0

    asm volatile("tensor_load_to_lds %0, %1" :: "s"(g0), "s"(g1) : "memory");
}

__global__ __launch_bounds__(256) void cube2equi_kernel(
    const float* __restrict__ x,      // [6*E, C, L, L]
    const int*   __restrict__ face,   // [H, W]
    const float* __restrict__ xy,     // [H, W, 2]
    float*       __restrict__ out)    // [E, C, H, W]
{
    constexpr int L  = CUBE_L, C = NCH, H = EQU_H, W = EQU_W;
    constexpr int LL = L * L;
    constexpr int HW = H * W;

    __shared__ int   s_face[8 * 32];   // 32x8 tile of face ids
    __shared__ float s_xy[8 * 64];     // 32x8 tile of (u,v) pairs, row = 64 floats

    const int tx = threadIdx.x;        // 0..31 (one wave32 per ty row)
    const int ty = threadIdx.y;        // 0..7
    const int bx = blockIdx.x * 32;
    const int by = blockIdx.y * 8;

    // Wave 0 stages the geometry maps for this tile into LDS via the TDM.
    if (ty == 0) {
        const uint64_t fga = (uint64_t)(uintptr_t)(face + (size_t)by * W + bx);
        const uint64_t xga = (uint64_t)(uintptr_t)(xy + ((size_t)by * W + bx) * 2);
        tdm_load_tile_2d((uint32_t)(uintptr_t)s_face, fga,
                         /*dims*/ W, H, /*stride*/ W, /*tile*/ 32, 8);
        tdm_load_tile_2d((uint32_t)(uintptr_t)s_xy, xga,
                         /*dims*/ 2 * W, H, /*stride*/ 2 * W, /*tile*/ 64, 8);
        __builtin_amdgcn_s_wait_tensorcnt(0);
    }
    __syncthreads();

    // Per-pixel geometry, computed once and reused for all 64 (e,c) images.
    const int   p = ty * 32 + tx;
    const int   f = s_face[p];
    const float u = s_xy[ty * 64 + tx * 2 + 0];
    const float v = s_xy[ty * 64 + tx * 2 + 1];

    int x0 = (int)floorf(u); x0 = x0 < 0 ? 0 : (x0 > L - 1 ? L - 1 : x0);
    int y0 = (int)floorf(v); y0 = y0 < 0 ? 0 : (y0 > L - 1 ? L - 1 : y0);
    const int x1 = (x0 + 1 > L - 1) ? L - 1 : x0 + 1;
    const int y1 = (y0 + 1 > L - 1) ? L - 1 : y0 + 1;

    const float wx  = u - (float)x0;
    const float wy  = v - (float)y0;
    const float w00 = (1.0f - wx) * (1.0f - wy);
    const float w01 = wx * (1.0f - wy);
    const float w10 = (1.0f - wx) * wy;
    const float w11 = wx * wy;

    const int i00 = y0 * L + x0;
    const int i01 = y0 * L + x1;
    const int i10 = y1 * L + x0;
    const int i11 = y1 * L + x1;

    // x layout: image (e,f,c) starts at ((e*6 + f)*C + c) * LL
    const float* b0 = x + (size_t)f * (C * LL);            // e=0, c=0
    const float* b1 = b0 + (size_t)6 * C * LL;             // e=1, c=0

    const int pix = (by + ty) * W + (bx + tx);
    float* o0 = out + pix;                                 // e=0, c=0
    float* o1 = out + (size_t)C * HW + pix;                // e=1, c=0

#pragma unroll 4
    for (int c = 0; c < C; ++c) {
        // Prefetch next channel's two tap rows (clamped in-bounds: WGP prefetch
        // is non-speculative on CDNA5, so never prefetch past the buffer).
        const int pf = (c < C - 1) ? LL : 0;
        __builtin_prefetch(b0 + pf + i00, 0, 0);
        __builtin_prefetch(b1 + pf + i10, 0, 0);

        const float s0 = w00 * b0[i00] + w01 * b0[i01] + w10 * b0[i10] + w11 * b0[i11];
        const float s1 = w00 * b1[i00] + w01 * b1[i01] + w10 * b1[i10] + w11 * b1[i11];

        // Output is write-once/streamed: NT stores keep L2 for the cube source.
        __builtin_nontemporal_store(s0, o0);
        __builtin_nontemporal_store(s1, o1);

        b0 += LL; b1 += LL;
        o0 += HW; o1 += HW;
    }
}

extern "C" void kernel_launch(void* const* d_in, const int* in_sizes, int n_in,
                              void* d_out, int out_size, void* d_ws, size_t ws_size,
                              hipStream_t stream) {
    const float* x    = (const float*)d_in[0];   // [12, 32, 512, 512] f32
    const int*   face = (const int*)d_in[1];     // [512, 1024] i32
    const float* xy   = (const float*)d_in[2];   // [512, 1024, 2] f32
    float* out = (float*)d_out;                  // [2, 32, 512, 1024] f32

    dim3 block(32, 8);
    dim3 grid(EQU_W / 32, EQU_H / 8);            // (32, 64)
    hipLaunchKernelGGL(cube2equi_kernel, grid, block, 0, stream, x, face, xy, out);
}